// TriangularLoss_22393959481614
// MI455X (gfx1250) — compile-verified
//
#include <hip/hip_runtime.h>

typedef float v2f __attribute__((ext_vector_type(2)));
typedef float v8f __attribute__((ext_vector_type(8)));

#define NROWS 8192
#define DIMS  256
#define NCLS  100
#define NT    (NROWS / 16)                 // 512 tiles per dimension
#define NTILES ((unsigned)(NT * (NT + 1) / 2))  // upper triangle incl. diagonal

// ---------------------------------------------------------------------------
// Kernel 1: per-row L2 norm -> invnorm[row], sq[row] (= ||fn_row||^2).
// Block 0 also zeroes the global class-pair accumulator S[100x100].
// One wave32 per row, 8 rows per 256-thread block.
// ---------------------------------------------------------------------------
__global__ __launch_bounds__(256) void prep_kernel(
    const float* __restrict__ feat, float* __restrict__ invn,
    float* __restrict__ sq, float* __restrict__ S) {
  const int lane = threadIdx.x & 31;
  const int wid  = threadIdx.x >> 5;
  const int row  = blockIdx.x * 8 + wid;
  const float* r = feat + (size_t)row * DIMS;
  float ss = 0.f;
#pragma unroll
  for (int c = 0; c < DIMS / 32; ++c) {
    float v = r[lane + 32 * c];
    ss += v * v;
  }
#pragma unroll
  for (int off = 16; off > 0; off >>= 1) ss += __shfl_xor(ss, off, 32);
  const float nrm = sqrtf(ss);
  const float iv  = 1.0f / fmaxf(nrm, 1e-12f);
  if (lane == 0) {
    invn[row] = iv;
    sq[row]   = ss * iv * iv;  // == sum(fn^2), ~1.0
  }
  if (blockIdx.x == 0)
    for (int i = threadIdx.x; i < NCLS * NCLS; i += 256) S[i] = 0.f;
}

// ---------------------------------------------------------------------------
// Kernel 2: persistent waves over upper-triangular 16x16 tile grid.
// FP32 WMMA (16x16x4) Gram tiles with TWO independent accumulator chains
// (breaks the serial D->C dependency), fused distance + class-pair scatter
// into per-workgroup LDS S, one global atomic flush per workgroup.
// ---------------------------------------------------------------------------
__global__ __launch_bounds__(256) void tile_kernel(
    const float* __restrict__ feat, const float* __restrict__ invn,
    const float* __restrict__ sq, const int* __restrict__ labels,
    float* __restrict__ S) {
  __shared__ float Sl[NCLS * NCLS];  // 40 KB
  for (int i = threadIdx.x; i < NCLS * NCLS; i += 256) Sl[i] = 0.f;
  __syncthreads();

  const int lane = threadIdx.x & 31;
  const int wid  = threadIdx.x >> 5;
  const int l15  = lane & 15;
  const int half = lane >> 4;  // 0: K{0,1}, 1: K{2,3}

  const unsigned stride = gridDim.x * 8u;  // total waves
  unsigned t = blockIdx.x * 8u + wid;

  // map linear index t -> (ti, tj), tj >= ti (row-major upper triangle)
  int ti = 0;
  unsigned rem = t;
  while (ti < NT && rem >= (unsigned)(NT - ti)) {
    rem -= (unsigned)(NT - ti);
    ++ti;
  }
  int tj = ti + (int)rem;

  for (; t < NTILES; t += stride) {
    const int i0 = ti * 16, j0 = tj * 16;
    // A frag: row = i0+l15, elems K = k + 2*half + {0,1}
    // B frag: B[k][n] = fn[j0+n][k] -> same access pattern on row j0+l15
    const float* Ar = feat + (size_t)(i0 + l15) * DIMS + 2 * half;
    const float* Br = feat + (size_t)(j0 + l15) * DIMS + 2 * half;

    v8f acc0 = {0.f, 0.f, 0.f, 0.f, 0.f, 0.f, 0.f, 0.f};
    v8f acc1 = {0.f, 0.f, 0.f, 0.f, 0.f, 0.f, 0.f, 0.f};
#pragma unroll 4
    for (int k = 0; k < DIMS; k += 8) {
      v2f a0 = *(const v2f*)(Ar + k);
      v2f b0 = *(const v2f*)(Br + k);
      v2f a1 = *(const v2f*)(Ar + k + 4);
      v2f b1 = *(const v2f*)(Br + k + 4);
      acc0 = __builtin_amdgcn_wmma_f32_16x16x4_f32(
          false, a0, false, b0, (short)0, acc0, false, false);
      acc1 = __builtin_amdgcn_wmma_f32_16x16x4_f32(
          false, a1, false, b1, (short)0, acc1, false, false);
    }

    // epilogue: D-matrix VGPR r holds M = r + 8*half, N = l15
    const int   j   = j0 + l15;
    const float ivj = invn[j];
    const float sqj = sq[j];
    const int   lb  = labels[j];
    const bool  offdiag = (tj > ti);
#pragma unroll
    for (int r = 0; r < 8; ++r) {
      const int   i  = i0 + r + 8 * half;
      const float g  = (acc0[r] + acc1[r]) * invn[i] * ivj;
      const float d2 = sq[i] + sqj - 2.0f * g;
      const float d  = (d2 > 0.f) ? __builtin_amdgcn_sqrtf(d2) : 0.f;
      const int   la = labels[i];
      atomicAdd(&Sl[la * NCLS + lb], d);
      if (offdiag) atomicAdd(&Sl[lb * NCLS + la], d);  // mirror (j,i)
    }

    // advance (ti, tj) by `stride` positions within the triangle
    tj += (int)stride;
    while (ti < NT && tj >= NT) {
      const int over = tj - NT;
      ++ti;
      tj = ti + over;
    }
  }

  __syncthreads();
  for (int i = threadIdx.x; i < NCLS * NCLS; i += 256) {
    const float v = Sl[i];
    if (v != 0.f) atomicAdd(&S[i], v);
  }
}

// ---------------------------------------------------------------------------
// Kernel 3: single block. Histogram labels, compute intra/inter means, loss.
// ---------------------------------------------------------------------------
__global__ __launch_bounds__(256) void finalize_kernel(
    const int* __restrict__ labels, const float* __restrict__ S,
    float* __restrict__ out) {
  __shared__ int   cnt[NCLS];
  __shared__ float facc[4];  // intraSum, intraN, interSum, interN
  const int tid = threadIdx.x;
  if (tid < NCLS) cnt[tid] = 0;
  if (tid < 4) facc[tid] = 0.f;
  __syncthreads();

  for (int i = tid; i < NROWS; i += 256) atomicAdd(&cnt[labels[i]], 1);
  __syncthreads();

  if (tid < NCLS) {
    const int c = cnt[tid];
    if (c > 1) {
      const float denom = (float)c * (float)(c - 1);
      atomicAdd(&facc[0], S[tid * NCLS + tid] / denom);
      atomicAdd(&facc[1], 1.f);
    }
  }
  for (int idx = tid; idx < NCLS * NCLS; idx += 256) {
    const int a = idx / NCLS, b = idx % NCLS;
    if (b > a) {
      const int ca = cnt[a], cb = cnt[b];
      if (ca > 0 && cb > 0) {
        atomicAdd(&facc[2], S[idx] / ((float)ca * (float)cb));
        atomicAdd(&facc[3], 1.f);
      }
    }
  }
  __syncthreads();
  if (tid == 0) {
    const float intra = (facc[1] > 0.f) ? facc[0] / facc[1] : 0.f;
    const float inter = (facc[3] > 0.f) ? facc[2] / facc[3] : 1.f;
    const float loss  = intra - inter + 1.0f;  // MARGIN = 1.0
    out[0] = (loss > 0.f) ? loss : 0.f;
  }
}

// ---------------------------------------------------------------------------
extern "C" void kernel_launch(void* const* d_in, const int* in_sizes, int n_in,
                              void* d_out, int out_size, void* d_ws,
                              size_t ws_size, hipStream_t stream) {
  (void)in_sizes; (void)n_in; (void)out_size; (void)ws_size;
  const float* feat   = (const float*)d_in[0];
  const int*   labels = (const int*)d_in[1];

  float* ws   = (float*)d_ws;
  float* invn = ws;                 // [8192]
  float* sq   = ws + NROWS;         // [8192]
  float* S    = ws + 2 * NROWS;     // [100*100]

  prep_kernel<<<NROWS / 8, 256, 0, stream>>>(feat, invn, sq, S);
  tile_kernel<<<512, 256, 0, stream>>>(feat, invn, sq, labels, S);
  finalize_kernel<<<1, 256, 0, stream>>>(labels, S, (float*)d_out);
}